// GATv2Block_781684048446
// MI455X (gfx1250) — compile-verified
//
#include <hip/hip_runtime.h>
#include <math.h>

#define N_NODES 50000
#define N_EDGES 800000
#define HID     128
#define EDIM    16
#define HEADS   4
#define FFN     512
#define NEG_SLOPE 0.2f
#define RMS_EPS   1.1920929e-07f   // FLT_EPSILON (torch RMSNorm eps=None)

typedef float v2f __attribute__((ext_vector_type(2)));
typedef float v8f __attribute__((ext_vector_type(8)));

// ---- fp32 WMMA: D = A(16x4) * B(4x16) + C(16x16) -----------------------
__device__ __forceinline__ v8f wmma_f32(v2f a, v2f b, v8f c) {
  return __builtin_amdgcn_wmma_f32_16x16x4_f32(false, a, false, b,
                                               (short)0, c, false, false);
}

// ========================================================================
// 0) init: degree/cursor/global-cursor = 0
// ========================================================================
__global__ void k_init(int* __restrict__ deg, int* __restrict__ cur,
                       int* __restrict__ gcur) {
  int i = blockIdx.x * blockDim.x + threadIdx.x;
  if (i < N_NODES) { deg[i] = 0; cur[i] = 0; }
  if (i == 0) *gcur = 0;
}

// ========================================================================
// 1) xl = x@Wl + bl ; xr = x@Wr + br   (WMMA fp32, shared A tile)
//    block = 256 thr (8 waves); tile = 16 rows x 128 cols; wave w -> cols 16w
// ========================================================================
__global__ __launch_bounds__(256) void k_node_gemm(
    const float* __restrict__ x,
    const float* __restrict__ Wl, const float* __restrict__ bl,
    const float* __restrict__ Wr, const float* __restrict__ br,
    float* __restrict__ xl, float* __restrict__ xr) {
  __shared__ float lds_a[16 * HID];
  const int tid  = threadIdx.x;
  const int wave = tid >> 5, lane = tid & 31;
  const int half = lane >> 4, m = lane & 15;
  const int r0   = blockIdx.x * 16;

  for (int i = tid; i < 16 * HID; i += 256)
    lds_a[i] = x[(size_t)(r0 + (i >> 7)) * HID + (i & 127)];
  __syncthreads();

  const int col0 = wave * 16;
  v8f cl, cr;
  const float bL = bl[col0 + m], bR = br[col0 + m];
  for (int v = 0; v < 8; ++v) { cl[v] = bL; cr[v] = bR; }

  for (int k = 0; k < HID; k += 4) {
    const int ka = k + 2 * half;
    v2f a, wl2, wr2;
    a[0]   = lds_a[m * HID + ka];
    a[1]   = lds_a[m * HID + ka + 1];
    wl2[0] = Wl[(size_t)ka * HID + col0 + m];
    wl2[1] = Wl[(size_t)(ka + 1) * HID + col0 + m];
    wr2[0] = Wr[(size_t)ka * HID + col0 + m];
    wr2[1] = Wr[(size_t)(ka + 1) * HID + col0 + m];
    cl = wmma_f32(a, wl2, cl);
    cr = wmma_f32(a, wr2, cr);
  }
  for (int v = 0; v < 8; ++v) {
    const size_t row = (size_t)(r0 + v + 8 * half);
    xl[row * HID + col0 + m] = cl[v];
    xr[row * HID + col0 + m] = cr[v];
  }
}

// ========================================================================
// 2) CSR build by destination: degree, segment alloc, placement
// ========================================================================
__global__ void k_degree(const int* __restrict__ ei, int* __restrict__ deg) {
  int e = blockIdx.x * blockDim.x + threadIdx.x;
  if (e >= N_EDGES) return;
  atomicAdd(&deg[ei[N_EDGES + e]], 1);
}

// segment order is irrelevant -> one atomicAdd on a global cursor, no scan
__global__ void k_alloc(const int* __restrict__ deg, int* __restrict__ rowstart,
                        int* __restrict__ gcur) {
  int n = blockIdx.x * blockDim.x + threadIdx.x;
  if (n >= N_NODES) return;
  rowstart[n] = atomicAdd(gcur, deg[n]);
}

__global__ void k_place(const int* __restrict__ ei,
                        const int* __restrict__ rowstart,
                        int* __restrict__ cur, int* __restrict__ eidlist) {
  int e = blockIdx.x * blockDim.x + threadIdx.x;
  if (e >= N_EDGES) return;
  const int d = ei[N_EDGES + e];
  const int slot = rowstart[d] + atomicAdd(&cur[d], 1);
  eidlist[slot] = e;
}

// ========================================================================
// 3) per-edge logits.  One wave per edge, lane = 4 channels.
//    m = xl[src] + xr[dst] + edge_attr@We ; LeakyReLU ; dot with att[head]
// ========================================================================
__global__ __launch_bounds__(256) void k_edge_logits(
    const int* __restrict__ ei, const float* __restrict__ ea,
    const float* __restrict__ We, const float* __restrict__ att,
    const float* __restrict__ xl, const float* __restrict__ xr,
    float* __restrict__ logits) {
  __shared__ float lds_we[EDIM * HID];   // 8 KB
  __shared__ float lds_att[HID];
  const int tid = threadIdx.x;
  for (int i = tid; i < EDIM * HID; i += 256) lds_we[i] = We[i];
  if (tid < HID) lds_att[tid] = att[tid];
  __syncthreads();

  const int lane   = tid & 31;
  const int h      = lane >> 3;                 // head = channel/32
  const int wgid   = blockIdx.x * 8 + (tid >> 5);
  const int nwaves = gridDim.x * 8;
  const float4 a4  = *(const float4*)&lds_att[lane * 4];

  for (int e = wgid; e < N_EDGES; e += nwaves) {
    const int s = ei[e], d = ei[N_EDGES + e];
    float4 m4 = ((const float4*)xl)[(size_t)s * 32 + lane];
    const float4 r4 = ((const float4*)xr)[(size_t)d * 32 + lane];
    m4.x += r4.x; m4.y += r4.y; m4.z += r4.z; m4.w += r4.w;

    const float4* eap = (const float4*)(ea + (size_t)e * EDIM);
#pragma unroll
    for (int k = 0; k < EDIM; k += 4) {
      const float4 ev = eap[k >> 2];
      const float* base = &lds_we[k * HID + lane * 4];
      const float4 w0 = *(const float4*)(base);
      const float4 w1 = *(const float4*)(base + HID);
      const float4 w2 = *(const float4*)(base + 2 * HID);
      const float4 w3 = *(const float4*)(base + 3 * HID);
      m4.x += ev.x * w0.x + ev.y * w1.x + ev.z * w2.x + ev.w * w3.x;
      m4.y += ev.x * w0.y + ev.y * w1.y + ev.z * w2.y + ev.w * w3.y;
      m4.z += ev.x * w0.z + ev.y * w1.z + ev.z * w2.z + ev.w * w3.z;
      m4.w += ev.x * w0.w + ev.y * w1.w + ev.z * w2.w + ev.w * w3.w;
    }
    const float sx = m4.x > 0.f ? m4.x : NEG_SLOPE * m4.x;
    const float sy = m4.y > 0.f ? m4.y : NEG_SLOPE * m4.y;
    const float sz = m4.z > 0.f ? m4.z : NEG_SLOPE * m4.z;
    const float sw = m4.w > 0.f ? m4.w : NEG_SLOPE * m4.w;
    float sum = sx * a4.x + sy * a4.y + sz * a4.z + sw * a4.w;
    sum += __shfl_xor(sum, 1, 32);
    sum += __shfl_xor(sum, 2, 32);
    sum += __shfl_xor(sum, 4, 32);                 // per-head (8 lanes) total
    if ((lane & 7) == 0)
      logits[(size_t)e * HEADS + h] = sum;
  }
}

// ========================================================================
// 4) node-centric softmax + aggregation + residual + RMSNorm-1.
//    One wave per node; gather-only, zero FP atomics.
// ========================================================================
__global__ __launch_bounds__(256) void k_node_agg_norm(
    const int* __restrict__ ei, const int* __restrict__ eidlist,
    const int* __restrict__ rowstart, const int* __restrict__ deg,
    const float* __restrict__ logits, const float* __restrict__ xl,
    const float* __restrict__ x, const float* __restrict__ bias_gat,
    const float* __restrict__ w1n, float* __restrict__ h) {
  const int lane = threadIdx.x & 31;
  const int node = blockIdx.x * 8 + (threadIdx.x >> 5);
  const int start = rowstart[node];
  const int cnt   = deg[node];
  const int hd    = lane >> 3;

  // ---- pass 1: per-head max over incoming logits (lane-strided) ----
  float4 mx; mx.x = mx.y = mx.z = mx.w = -INFINITY;
  for (int i = lane; i < cnt; i += 32) {
    const int eid = eidlist[start + i];
    const float4 lg = *(const float4*)&logits[(size_t)eid * HEADS];
    mx.x = fmaxf(mx.x, lg.x); mx.y = fmaxf(mx.y, lg.y);
    mx.z = fmaxf(mx.z, lg.z); mx.w = fmaxf(mx.w, lg.w);
  }
  for (int o = 1; o < 32; o <<= 1) {
    mx.x = fmaxf(mx.x, __shfl_xor(mx.x, o, 32));
    mx.y = fmaxf(mx.y, __shfl_xor(mx.y, o, 32));
    mx.z = fmaxf(mx.z, __shfl_xor(mx.z, o, 32));
    mx.w = fmaxf(mx.w, __shfl_xor(mx.w, o, 32));
  }

  // ---- pass 2: per-head exp-sum ----
  float4 dn; dn.x = dn.y = dn.z = dn.w = 0.f;
  for (int i = lane; i < cnt; i += 32) {
    const int eid = eidlist[start + i];
    const float4 lg = *(const float4*)&logits[(size_t)eid * HEADS];
    dn.x += expf(lg.x - mx.x); dn.y += expf(lg.y - mx.y);
    dn.z += expf(lg.z - mx.z); dn.w += expf(lg.w - mx.w);
  }
  for (int o = 1; o < 32; o <<= 1) {
    dn.x += __shfl_xor(dn.x, o, 32);
    dn.y += __shfl_xor(dn.y, o, 32);
    dn.z += __shfl_xor(dn.z, o, 32);
    dn.w += __shfl_xor(dn.w, o, 32);
  }

  // this lane's head values
  const float mh = (hd & 2) ? ((hd & 1) ? mx.w : mx.z)
                            : ((hd & 1) ? mx.y : mx.x);
  const float dh = (hd & 2) ? ((hd & 1) ? dn.w : dn.z)
                            : ((hd & 1) ? dn.y : dn.x);
  const float inv = 1.0f / (dh + 1e-16f);

  // ---- pass 3: weighted gather of xl[src] (all lanes per edge) ----
  float4 acc; acc.x = acc.y = acc.z = acc.w = 0.f;
  for (int i = 0; i < cnt; ++i) {
    const int eid = eidlist[start + i];
    const int s   = ei[eid];
    const float4 lg = *(const float4*)&logits[(size_t)eid * HEADS];
    const float lgh = (hd & 2) ? ((hd & 1) ? lg.w : lg.z)
                               : ((hd & 1) ? lg.y : lg.x);
    const float alpha = expf(lgh - mh) * inv;
    const float4 v = ((const float4*)xl)[(size_t)s * 32 + lane];
    acc.x += alpha * v.x; acc.y += alpha * v.y;
    acc.z += alpha * v.z; acc.w += alpha * v.w;
  }

  // ---- residual + rmsnorm(x + gat + bias_gat, w_norm1) ----
  const float4 xv = ((const float4*)x)[(size_t)node * 32 + lane];
  const float4 bv = ((const float4*)bias_gat)[lane];
  float4 v;
  v.x = xv.x + acc.x + bv.x; v.y = xv.y + acc.y + bv.y;
  v.z = xv.z + acc.z + bv.z; v.w = xv.w + acc.w + bv.w;
  float ss = v.x * v.x + v.y * v.y + v.z * v.z + v.w * v.w;
  for (int o = 1; o < 32; o <<= 1) ss += __shfl_xor(ss, o, 32);
  const float sc = rsqrtf(ss * (1.0f / HID) + RMS_EPS);
  const float4 w = ((const float4*)w1n)[lane];
  float4 o4;
  o4.x = v.x * sc * w.x; o4.y = v.y * sc * w.y;
  o4.z = v.z * sc * w.z; o4.w = v.w * sc * w.w;
  ((float4*)h)[(size_t)node * 32 + lane] = o4;
}

// ========================================================================
// 5) f1 = gelu_exact(h @ W1 + b1)   tile 16 rows x 512 cols per block
// ========================================================================
__global__ __launch_bounds__(256) void k_ffn1(
    const float* __restrict__ h, const float* __restrict__ W1,
    const float* __restrict__ b1, float* __restrict__ f1) {
  __shared__ float lds_a[16 * HID];
  const int tid = threadIdx.x, wave = tid >> 5, lane = tid & 31;
  const int half = lane >> 4, m = lane & 15;
  const int r0 = blockIdx.x * 16;

  for (int i = tid; i < 16 * HID; i += 256)
    lds_a[i] = h[(size_t)(r0 + (i >> 7)) * HID + (i & 127)];
  __syncthreads();

  for (int nt = 0; nt < 4; ++nt) {
    const int col0 = wave * 64 + nt * 16;
    v8f c;
    const float bb = b1[col0 + m];
    for (int v = 0; v < 8; ++v) c[v] = bb;
    for (int k = 0; k < HID; k += 4) {
      const int ka = k + 2 * half;
      v2f a, b;
      a[0] = lds_a[m * HID + ka];
      a[1] = lds_a[m * HID + ka + 1];
      b[0] = W1[(size_t)ka * FFN + col0 + m];
      b[1] = W1[(size_t)(ka + 1) * FFN + col0 + m];
      c = wmma_f32(a, b, c);
    }
    for (int v = 0; v < 8; ++v) {
      float val = c[v];
      val = 0.5f * val * (1.0f + erff(val * 0.70710678118654752f));
      f1[(size_t)(r0 + v + 8 * half) * FFN + col0 + m] = val;
    }
  }
}

// ========================================================================
// 6) out = rmsnorm(h + f1 @ W2 + b2, w_norm2)  fused per 16-row tile
// ========================================================================
__global__ __launch_bounds__(256) void k_ffn2_norm(
    const float* __restrict__ f1, const float* __restrict__ W2,
    const float* __restrict__ b2, const float* __restrict__ h,
    const float* __restrict__ w2n, float* __restrict__ out) {
  __shared__ float smem[16 * FFN];                 // 32 KB, reused as out tile
  const int tid = threadIdx.x, wave = tid >> 5, lane = tid & 31;
  const int half = lane >> 4, m = lane & 15;
  const int r0 = blockIdx.x * 16;

  for (int i = tid; i < 16 * FFN; i += 256)
    smem[i] = f1[(size_t)(r0 + (i >> 9)) * FFN + (i & 511)];
  __syncthreads();

  const int col0 = wave * 16;
  v8f c;
  const float bb = b2[col0 + m];
  for (int v = 0; v < 8; ++v) c[v] = bb;
  for (int k = 0; k < FFN; k += 4) {
    const int ka = k + 2 * half;
    v2f a, b;
    a[0] = smem[m * FFN + ka];
    a[1] = smem[m * FFN + ka + 1];
    b[0] = W2[(size_t)ka * HID + col0 + m];
    b[1] = W2[(size_t)(ka + 1) * HID + col0 + m];
    c = wmma_f32(a, b, c);
  }
  __syncthreads();                                 // done reading A tile
  for (int v = 0; v < 8; ++v) {
    const int row = v + 8 * half;
    const float hv = h[(size_t)(r0 + row) * HID + col0 + m];
    smem[row * HID + col0 + m] = c[v] + hv;        // residual
  }
  __syncthreads();
  for (int rr = 0; rr < 2; ++rr) {                 // wave -> 2 rows
    const int row = wave * 2 + rr;
    const float4 v4 = *(const float4*)&smem[row * HID + lane * 4];
    float ss = v4.x * v4.x + v4.y * v4.y + v4.z * v4.z + v4.w * v4.w;
    for (int o = 1; o < 32; o <<= 1) ss += __shfl_xor(ss, o, 32);
    const float sc = rsqrtf(ss * (1.0f / HID) + RMS_EPS);
    const float4 w = ((const float4*)w2n)[lane];
    float4 o4;
    o4.x = v4.x * sc * w.x; o4.y = v4.y * sc * w.y;
    o4.z = v4.z * sc * w.z; o4.w = v4.w * sc * w.w;
    *(float4*)&out[(size_t)(r0 + row) * HID + lane * 4] = o4;
  }
}

// ========================================================================
extern "C" void kernel_launch(void* const* d_in, const int* in_sizes, int n_in,
                              void* d_out, int out_size, void* d_ws, size_t ws_size,
                              hipStream_t stream) {
  (void)in_sizes; (void)n_in; (void)out_size; (void)ws_size;
  const float* x        = (const float*)d_in[0];
  const int*   ei       = (const int*)d_in[1];
  const float* ea       = (const float*)d_in[2];
  const float* Wl       = (const float*)d_in[3];
  const float* bl       = (const float*)d_in[4];
  const float* Wr       = (const float*)d_in[5];
  const float* br       = (const float*)d_in[6];
  const float* We       = (const float*)d_in[7];
  const float* att      = (const float*)d_in[8];
  const float* bias_gat = (const float*)d_in[9];
  const float* w1n      = (const float*)d_in[10];
  const float* w2n      = (const float*)d_in[11];
  const float* fw1      = (const float*)d_in[12];
  const float* fb1      = (const float*)d_in[13];
  const float* fw2      = (const float*)d_in[14];
  const float* fb2      = (const float*)d_in[15];
  float* out = (float*)d_out;

  float* ws     = (float*)d_ws;
  float* xl     = ws;                               // N*128
  float* xr     = xl + (size_t)N_NODES * HID;       // N*128
  float* hbuf   = xr + (size_t)N_NODES * HID;       // N*128
  float* logits = hbuf + (size_t)N_NODES * HID;     // E*4
  float* f1     = logits + (size_t)N_EDGES * HEADS; // N*512
  int*   ideg   = (int*)(f1 + (size_t)N_NODES * FFN);
  int*   irow   = ideg + N_NODES;
  int*   icur   = irow + N_NODES;
  int*   igcur  = icur + N_NODES;                   // 1
  int*   ielist = igcur + 1;                        // E

  k_init<<<(N_NODES + 255) / 256, 256, 0, stream>>>(ideg, icur, igcur);
  k_node_gemm<<<N_NODES / 16, 256, 0, stream>>>(x, Wl, bl, Wr, br, xl, xr);
  k_degree<<<(N_EDGES + 255) / 256, 256, 0, stream>>>(ei, ideg);
  k_alloc<<<(N_NODES + 255) / 256, 256, 0, stream>>>(ideg, irow, igcur);
  k_place<<<(N_EDGES + 255) / 256, 256, 0, stream>>>(ei, irow, icur, ielist);
  k_edge_logits<<<4096, 256, 0, stream>>>(ei, ea, We, att, xl, xr, logits);
  k_node_agg_norm<<<N_NODES / 8, 256, 0, stream>>>(ei, ielist, irow, ideg,
                                                   logits, xl, x, bias_gat,
                                                   w1n, hbuf);
  k_ffn1<<<N_NODES / 16, 256, 0, stream>>>(hbuf, fw1, fb1, f1);
  k_ffn2_norm<<<N_NODES / 16, 256, 0, stream>>>(f1, fw2, fb2, hbuf, w2n, out);
}